// Erosion2D_82231443849912
// MI455X (gfx1250) — compile-verified
//
#include <hip/hip_runtime.h>
#include <stdint.h>

// ---------------------------------------------------------------------------
// erosion2d (TF SAME, stride 1, 4x4 kernel, NHWC 8x512x512x32, fp32)
//   out[b,y,x,c] = min_{dy,dx in [0,4)} X[b, y+dy-1, x+dx-1, c] - w[3-dy,3-dx,c]
//   (out-of-range X == +1e30 padding; real taps always dominate the min, so
//    filling the halo with 1e30 is bit-exact vs. the padded reference)
//
// MI455X strategy (bandwidth-bound: ~512MB traffic, 23.3TB/s -> ~22us floor):
//   * wave32 lane == channel (C==32): every access is a coalesced 128B vector
//   * stage 11x35-pixel halo tile into LDS once via gfx1250 async
//     GLOBAL_LOAD_ASYNC_TO_LDS_B128 (ASYNCcnt) + s_wait_asynccnt
//   * column-ring register reuse: 4 LDS loads/output instead of 16
//   * non-temporal output stores (write-once data; preserve L2 for halo reuse)
// ---------------------------------------------------------------------------

#define EH   512
#define EW   512
#define EC   32
#define BIGF 1e30f

#define TILE_W 32
#define TILE_H 8
#define LROWS  (TILE_H + 3)            // 11 input rows
#define LCOLS  (TILE_W + 3)            // 35 input pixel columns
#define LROWF  (LCOLS * EC)            // 1120 floats per LDS row
#define CHUNKS_PER_ROW (LCOLS * 8)     // 280 x 16B chunks per row
#define TOTAL_CHUNKS   (LROWS * CHUNKS_PER_ROW)  // 3080

#if __has_builtin(__builtin_amdgcn_global_load_async_to_lds_b128)
#define USE_ASYNC_LDS 1
#endif

typedef int v4i __attribute__((__vector_size__(4 * sizeof(int))));
typedef __attribute__((address_space(1))) v4i v4i_g;   // global
typedef __attribute__((address_space(3))) v4i v4i_l;   // LDS

#if __has_builtin(__builtin_amdgcn_s_wait_asynccnt)
#define WAIT_ASYNCCNT0() __builtin_amdgcn_s_wait_asynccnt(0)
#else
#define WAIT_ASYNCCNT0() asm volatile("s_wait_asynccnt 0" ::: "memory")
#endif

#if __has_builtin(__builtin_nontemporal_store)
#define NT_STORE(val, ptr) __builtin_nontemporal_store((val), (ptr))
#else
#define NT_STORE(val, ptr) (*(ptr) = (val))
#endif

__global__ __launch_bounds__(256) void erosion2d_kernel(
    const float* __restrict__ xg,   // [8,512,512,32]
    const float* __restrict__ wg,   // [4,4,32]
    float* __restrict__ og)         // [8,512,512,32]
{
    __shared__ float lds[LROWS * LROWF];   // 49,280 bytes

    const int tid   = threadIdx.x;
    const int tileX = blockIdx.x;          // 0..15
    const int tileY = blockIdx.y;          // 0..63
    const int b     = blockIdx.z;          // 0..7

    const int gx0 = tileX * TILE_W - 1;    // leftmost input column (may be -1)
    const int gy0 = tileY * TILE_H - 1;    // topmost input row (may be -1)
    const size_t batchBase = (size_t)b * EH * EW * EC;

    // ---- stage input halo tile (11 rows x 35 px x 32 ch) into LDS ----------
    for (int i = tid; i < TOTAL_CHUNKS; i += 256) {
        const int r  = i / CHUNKS_PER_ROW;
        const int k  = i - r * CHUNKS_PER_ROW;   // 16B chunk within row
        const int y  = gy0 + r;
        const int px = gx0 + (k >> 3);           // pixel column of this chunk
        float* dst = &lds[r * LROWF + k * 4];    // chunk k -> floats [4k,4k+4)
        if ((unsigned)y < (unsigned)EH && (unsigned)px < (unsigned)EW) {
            const float* src =
                xg + batchBase + ((size_t)y * EW + px) * EC + (k & 7) * 4;
#if defined(USE_ASYNC_LDS)
            // GLOBAL_LOAD_ASYNC_TO_LDS_B128: per-lane 16B, tracked by ASYNCcnt
            __builtin_amdgcn_global_load_async_to_lds_b128(
                (v4i_g*)src, (v4i_l*)dst, /*offset=*/0, /*cpol=*/0);
#else
            dst[0] = src[0]; dst[1] = src[1]; dst[2] = src[2]; dst[3] = src[3];
#endif
        } else {
            // +inf padding stand-in (exactly dominates: real taps always win)
            dst[0] = BIGF; dst[1] = BIGF; dst[2] = BIGF; dst[3] = BIGF;
        }
    }
#if defined(USE_ASYNC_LDS)
    WAIT_ASYNCCNT0();                      // drain async LDS writes
#endif
    __syncthreads();

    // ---- compute: wave wv -> output row, lane -> channel --------------------
    const int wv = tid >> 5;               // 0..7 (output row within tile)
    const int c  = tid & 31;               // channel

    // reflected structuring element, per-lane: wr[dy*4+dx] = w[3-dy][3-dx][c]
    float wr[16];
#pragma unroll
    for (int dy = 0; dy < 4; ++dy)
#pragma unroll
        for (int dx = 0; dx < 4; ++dx)
            wr[dy * 4 + dx] = wg[((3 - dy) * 4 + (3 - dx)) * EC + c];

    const int yo = tileY * TILE_H + wv;
    float* orow = og + batchBase + ((size_t)yo * EW + tileX * TILE_W) * EC + c;

    // Column-ring: keep 4 input columns x 4 rows in registers.
    // LDS row (wv+dy) <-> input row (yo+dy-1); LDS col j <-> input col gx0+j.
    // Output xo uses LDS cols xo..xo+3 (dx = 0..3).
    float A[4], B[4], C[4], D[4];

#define LOAD_COL(dstv, j)                                        \
    {                                                            \
        _Pragma("unroll")                                        \
        for (int dy = 0; dy < 4; ++dy)                           \
            dstv[dy] = lds[(wv + dy) * LROWF + (j) * EC + c];    \
    }

#define EMIT(xo, C0, C1, C2, C3)                                 \
    {                                                            \
        float m = BIGF;                                          \
        _Pragma("unroll")                                        \
        for (int dy = 0; dy < 4; ++dy) {                         \
            m = fminf(m, C0[dy] - wr[dy * 4 + 0]);               \
            m = fminf(m, C1[dy] - wr[dy * 4 + 1]);               \
            m = fminf(m, C2[dy] - wr[dy * 4 + 2]);               \
            m = fminf(m, C3[dy] - wr[dy * 4 + 3]);               \
        }                                                        \
        NT_STORE(m, &orow[(size_t)(xo) * EC]);                   \
    }

    LOAD_COL(A, 0)
    LOAD_COL(B, 1)
    LOAD_COL(C, 2)
#pragma unroll
    for (int k = 0; k < 8; ++k) {
        LOAD_COL(D, 4 * k + 3) EMIT(4 * k + 0, A, B, C, D)
        LOAD_COL(A, 4 * k + 4) EMIT(4 * k + 1, B, C, D, A)
        LOAD_COL(B, 4 * k + 5) EMIT(4 * k + 2, C, D, A, B)
        LOAD_COL(C, 4 * k + 6) EMIT(4 * k + 3, D, A, B, C)
    }
#undef LOAD_COL
#undef EMIT
}

extern "C" void kernel_launch(void* const* d_in, const int* in_sizes, int n_in,
                              void* d_out, int out_size, void* d_ws, size_t ws_size,
                              hipStream_t stream) {
    (void)in_sizes; (void)n_in; (void)out_size; (void)d_ws; (void)ws_size;
    const float* x = (const float*)d_in[0];   // [8,512,512,32] fp32
    const float* w = (const float*)d_in[1];   // [4,4,32] fp32
    float* out = (float*)d_out;               // [8,512,512,32] fp32

    dim3 grid(EW / TILE_W, EH / TILE_H, 8);   // (16, 64, 8)
    erosion2d_kernel<<<grid, 256, 0, stream>>>(x, w, out);
}